// DAM_Module_62457414418938
// MI455X (gfx1250) — compile-verified
//
#include <hip/hip_runtime.h>
#include <hip/hip_bf16.h>
#include <stdint.h>

typedef __attribute__((ext_vector_type(2))) float v2f;
typedef __attribute__((ext_vector_type(8))) float v8f;

// Problem constants
#define BB     2
#define CC     64
#define NN     8192          // L*H*W = 8*32*32
#define TC     192           // 3*C

// ---------------------------------------------------------------------------
// Kernel 1: produce pq[b,n], pk[b,n], pv[b,c,n]
//   pq = <q[:,n], wq> + bq ; pk likewise ; pv[c,n] = bv[c] + sum_d wv[c,d]*val[d,n]
//   val = concat(v, q, k) over channels.
// ---------------------------------------------------------------------------
#define NCH_A  64
#define PA     68            // LDS row pitch (floats): 68*4B, 16B-aligned rows

__global__ __launch_bounds__(256) void dam_pre_kernel(
    const float* __restrict__ q, const float* __restrict__ k,
    const float* __restrict__ v,
    const float* __restrict__ wq, const float* __restrict__ bq,
    const float* __restrict__ wk, const float* __restrict__ bk,
    const float* __restrict__ wv, const float* __restrict__ bv,
    float* __restrict__ pq, float* __restrict__ pk, float* __restrict__ pvout)
{
    __shared__ float vs[CC * PA];
    __shared__ float qs[CC * PA];
    __shared__ float ks[CC * PA];

    const int b   = blockIdx.y;
    const int n0  = blockIdx.x * NCH_A;
    const int tid = threadIdx.x;

    const float* qb = q + b * CC * NN;
    const float* kb = k + b * CC * NN;
    const float* vb = v + b * CC * NN;

#pragma unroll
    for (int i = 0; i < 4; ++i) {
        int idx4 = tid + 256 * i;            // 0..1023
        int c    = idx4 >> 4;                // 16 float4 per row
        int nin  = (idx4 & 15) << 2;
        *(float4*)&qs[c * PA + nin] = *(const float4*)(qb + c * NN + n0 + nin);
        *(float4*)&ks[c * PA + nin] = *(const float4*)(kb + c * NN + n0 + nin);
        *(float4*)&vs[c * PA + nin] = *(const float4*)(vb + c * NN + n0 + nin);
    }
    __syncthreads();

    if (tid < NCH_A) {
        float aq = bq[0], ak = bk[0];
#pragma unroll 8
        for (int c = 0; c < CC; ++c) {
            aq += qs[c * PA + tid] * wq[c];
            ak += ks[c * PA + tid] * wk[c];
        }
        pq[b * NN + n0 + tid] = aq;
        pk[b * NN + n0 + tid] = ak;
    }

    const int nloc = tid & 63;
    const int cg   = tid >> 6;               // 0..3, uniform per wave
    float acc[16];
#pragma unroll
    for (int cc = 0; cc < 16; ++cc) acc[cc] = bv[cg * 16 + cc];

    for (int d = 0; d < CC; ++d) {
        float xv = vs[d * PA + nloc];
        float xq = qs[d * PA + nloc];
        float xk = ks[d * PA + nloc];
#pragma unroll
        for (int cc = 0; cc < 16; ++cc) {
            const float* wr = wv + (cg * 16 + cc) * TC;
            acc[cc] += wr[d] * xv + wr[64 + d] * xq + wr[128 + d] * xk;
        }
    }
#pragma unroll
    for (int cc = 0; cc < 16; ++cc)
        pvout[b * CC * NN + (cg * 16 + cc) * NN + n0 + nloc] = acc[cc];
}

// ---------------------------------------------------------------------------
// Kernel 2: per-batch max/min of pk (exact softmax row-max for rank-1 energy)
// ---------------------------------------------------------------------------
__global__ __launch_bounds__(256) void dam_minmax_kernel(
    const float* __restrict__ pk, float* __restrict__ pkmax, float* __restrict__ pkmin)
{
    __shared__ float smax[256];
    __shared__ float smin[256];
    const int b = blockIdx.x;
    const int t = threadIdx.x;
    float mx = -3.402823466e38f, mn = 3.402823466e38f;
    for (int n = t; n < NN; n += 256) {
        float x = pk[b * NN + n];
        mx = fmaxf(mx, x);
        mn = fminf(mn, x);
    }
    smax[t] = mx; smin[t] = mn;
    __syncthreads();
    for (int s = 128; s > 0; s >>= 1) {
        if (t < s) {
            smax[t] = fmaxf(smax[t], smax[t + s]);
            smin[t] = fminf(smin[t], smin[t + s]);
        }
        __syncthreads();
    }
    if (t == 0) { pkmax[b] = smax[0]; pkmin[b] = smin[0]; }
}

// ---------------------------------------------------------------------------
// Kernel 3: fused exp + fp32 WMMA GEMM + softmax-normalize epilogue.
//   out[b,c,m] = gamma * (sum_n pv[c,n]*w[m,n]) / (sum_n w[m,n]) + v[b,c,m]
//   w[m,n]    = exp2(pq2[m]*pk[n] - rm2[m])   (log2e pre-folded into pq/rowmax)
//
// Block: 256 threads = 8 waves, m-tile 128 per block; all waves share the LDS
// pv chunk. pv/pk chunks are streamed global->LDS with CDNA5 async-to-LDS
// (ASYNCcnt) into a double buffer, overlapping the WMMA K-loop of the current
// chunk. A fragments (16x4 exp weights) are built in-register; their per-lane
// running sum gives the softmax denominator for free.
// ---------------------------------------------------------------------------
#define NCH_C  64
#define PC     68            // pitch % 64 == 4 -> conflict-free ds_load_b64

__global__ __launch_bounds__(256) void dam_attn_kernel(
    const float* __restrict__ pq, const float* __restrict__ pk,
    const float* __restrict__ pv,
    const float* __restrict__ pkmax, const float* __restrict__ pkmin,
    const float* __restrict__ vin, const float* __restrict__ gamma,
    float* __restrict__ out)
{
    __shared__ float pvs[2][CC * PC];        // 2 x 17408 B
    __shared__ float pks[2][80];

    const int b    = blockIdx.y;
    const int tid  = threadIdx.x;
    const int lane = tid & 31;
    const int wave = tid >> 5;
    const int half = lane >> 4;              // 0: lanes 0-15, 1: lanes 16-31
    const int l16  = lane & 15;
    const int mbase = blockIdx.x * 128 + wave * 16;

    const float LOG2E  = 1.4426950408889634f;
    const float pqm    = pq[b * NN + mbase + l16];
    const float rowmax = (pqm >= 0.0f) ? pqm * pkmax[b] : pqm * pkmin[b];
    const float pqm2   = pqm * LOG2E;
    const float rm2    = rowmax * LOG2E;

    const float* pvb = pv + b * CC * NN;
    const float* pkb = pk + b * NN;

    // Issue async global->LDS copies for one 64-wide n-chunk (ASYNCcnt).
    auto issue_chunk = [&](int n0, int bsel) {
#pragma unroll
        for (int i = 0; i < 4; ++i) {
            int idx4 = tid + 256 * i;        // 0..1023 float4s (64c x 16/row)
            int c    = idx4 >> 4;
            int nin  = (idx4 & 15) << 2;
            unsigned lds = (unsigned)(uintptr_t)&pvs[bsel][c * PC + nin];
            unsigned long long ga =
                (unsigned long long)(uintptr_t)(pvb + c * NN + n0 + nin);
            asm volatile("global_load_async_to_lds_b128 %0, %1, off"
                         :: "v"(lds), "v"(ga) : "memory");
        }
        if (tid < 16) {
            unsigned lds = (unsigned)(uintptr_t)&pks[bsel][tid * 4];
            unsigned long long ga =
                (unsigned long long)(uintptr_t)(pkb + n0 + tid * 4);
            asm volatile("global_load_async_to_lds_b128 %0, %1, off"
                         :: "v"(lds), "v"(ga) : "memory");
        }
    };

    v8f acc[4];
#pragma unroll
    for (int ct = 0; ct < 4; ++ct) acc[ct] = (v8f)(0.0f);
    float dacc = 0.0f;

    issue_chunk(0, 0);
    asm volatile("s_wait_asynccnt 0x0" ::: "memory");
    __syncthreads();

    const int NCHUNKS = NN / NCH_C;          // 128
    for (int ci = 0; ci < NCHUNKS; ++ci) {
        const int bsel = ci & 1;
        if (ci + 1 < NCHUNKS) issue_chunk((ci + 1) * NCH_C, bsel ^ 1);

        const float* pvloc = &pvs[bsel][0];
        const float* pkloc = &pks[bsel][0];
#pragma unroll 4
        for (int kk = 0; kk < NCH_C; kk += 4) {
            // A: lane holds w[m=l16, kk+2*half], w[m, kk+2*half+1]
            float2 pkp = *(const float2*)&pkloc[kk + 2 * half];
            v2f a;
            a.x = __builtin_amdgcn_exp2f(__builtin_fmaf(pqm2, pkp.x, -rm2));
            a.y = __builtin_amdgcn_exp2f(__builtin_fmaf(pqm2, pkp.y, -rm2));
            dacc += a.x + a.y;               // softmax denominator (partial)

            const float* bp = &pvloc[l16 * PC + kk + 2 * half];
            v2f b0 = *(const v2f*)(bp);
            v2f b1 = *(const v2f*)(bp + 16 * PC);
            v2f b2 = *(const v2f*)(bp + 32 * PC);
            v2f b3 = *(const v2f*)(bp + 48 * PC);

            acc[0] = __builtin_amdgcn_wmma_f32_16x16x4_f32(false, a, false, b0, (short)0, acc[0], false, false);
            acc[1] = __builtin_amdgcn_wmma_f32_16x16x4_f32(false, a, false, b1, (short)0, acc[1], false, false);
            acc[2] = __builtin_amdgcn_wmma_f32_16x16x4_f32(false, a, false, b2, (short)0, acc[2], false, false);
            acc[3] = __builtin_amdgcn_wmma_f32_16x16x4_f32(false, a, false, b3, (short)0, acc[3], false, false);
        }

        if (ci + 1 < NCHUNKS) {
            asm volatile("s_wait_asynccnt 0x0" ::: "memory");
            __syncthreads();
        }
    }

    // denominator: combine the two K-halves; lane L then holds den[m = L&15]
    dacc += __shfl_xor(dacc, 16, 32);
    float rd[8];
#pragma unroll
    for (int r = 0; r < 8; ++r)
        rd[r] = 1.0f / __shfl(dacc, r + 8 * half, 32);   // den for D-row r

    const float g  = gamma[0];
    const int   m0 = mbase + 8 * half;       // D VGPR r -> m = m0 + r
    const float* vb = vin + b * CC * NN;
    float*      ob = out + b * CC * NN;

#pragma unroll
    for (int ct = 0; ct < 4; ++ct) {
        int c = ct * 16 + l16;
        const float* vp = vb + c * NN + m0;
        float*       op = ob + c * NN + m0;
        float4 va = *(const float4*)(vp);
        float4 vc = *(const float4*)(vp + 4);
        float4 o0, o1;
        o0.x = g * acc[ct][0] * rd[0] + va.x;
        o0.y = g * acc[ct][1] * rd[1] + va.y;
        o0.z = g * acc[ct][2] * rd[2] + va.z;
        o0.w = g * acc[ct][3] * rd[3] + va.w;
        o1.x = g * acc[ct][4] * rd[4] + vc.x;
        o1.y = g * acc[ct][5] * rd[5] + vc.y;
        o1.z = g * acc[ct][6] * rd[6] + vc.z;
        o1.w = g * acc[ct][7] * rd[7] + vc.w;
        *(float4*)(op)     = o0;
        *(float4*)(op + 4) = o1;
    }
}

// ---------------------------------------------------------------------------
// Workspace layout (floats): pq[B*N] | pk[B*N] | pv[B*C*N] | pkmax[B] | pkmin[B]
// total ~4.33 MB
// ---------------------------------------------------------------------------
extern "C" void kernel_launch(void* const* d_in, const int* in_sizes, int n_in,
                              void* d_out, int out_size, void* d_ws, size_t ws_size,
                              hipStream_t stream)
{
    const float* q  = (const float*)d_in[0];
    const float* k  = (const float*)d_in[1];
    const float* v  = (const float*)d_in[2];
    const float* wq = (const float*)d_in[3];
    const float* bq = (const float*)d_in[4];
    const float* wk = (const float*)d_in[5];
    const float* bk = (const float*)d_in[6];
    const float* wv = (const float*)d_in[7];
    const float* bv = (const float*)d_in[8];
    const float* gx = (const float*)d_in[9];

    float* ws    = (float*)d_ws;
    float* pq    = ws;
    float* pk    = ws + BB * NN;
    float* pv    = ws + 2 * BB * NN;
    float* pkmax = pv + BB * CC * NN;
    float* pkmin = pkmax + BB;

    dam_pre_kernel<<<dim3(NN / NCH_A, BB), 256, 0, stream>>>(
        q, k, v, wq, bq, wk, bk, wv, bv, pq, pk, pv);
    dam_minmax_kernel<<<dim3(BB), 256, 0, stream>>>(pk, pkmax, pkmin);
    dam_attn_kernel<<<dim3(NN / 128, BB), 256, 0, stream>>>(
        pq, pk, pv, pkmax, pkmin, v, gx, (float*)d_out);
}